// GCN_41042707481217
// MI455X (gfx1250) — compile-verified
//
#include <hip/hip_runtime.h>
#include <hip/hip_bf16.h>

typedef __attribute__((ext_vector_type(2))) float v2f;
typedef __attribute__((ext_vector_type(8))) float v8f;
typedef __attribute__((ext_vector_type(4))) int   v4i;

#define FDIM 128

#define AS_GLOBAL __attribute__((address_space(1)))
#define AS_SHARED __attribute__((address_space(3)))

// Async global->LDS helper (CDNA5 GLOBAL_LOAD_ASYNC_TO_LDS_B128, ASYNCcnt
// tracked). Builtin signature (from hipcc diagnostic): (AS1 v4i*, AS3 v4i*,
// imm offset, imm cpol). Falls back to plain load+store if absent.
__device__ __forceinline__ void async_copy_b128_to_lds(float* lds_dst, const float* gsrc)
{
#if __has_builtin(__builtin_amdgcn_global_load_async_to_lds_b128)
    __builtin_amdgcn_global_load_async_to_lds_b128(
        (AS_GLOBAL v4i*)gsrc, (AS_SHARED v4i*)lds_dst, 0, 0);
#else
    *(float4*)lds_dst = *(const float4*)gsrc;
#endif
}

__device__ __forceinline__ void async_copy_wait()
{
#if __has_builtin(__builtin_amdgcn_global_load_async_to_lds_b128)
#if __has_builtin(__builtin_amdgcn_s_wait_asynccnt)
    __builtin_amdgcn_s_wait_asynccnt(0);
#else
    asm volatile("s_wait_asynccnt 0x0" ::: "memory");
#endif
#endif
}

// ---------------------------------------------------------------------------
// WMMA f32 GEMM: C[M,COLS] = A[M,K] @ B[K,COLS] (+bias) (+ReLU)
// One wave computes a 16x16 tile via V_WMMA_F32_16X16X4_F32; the block's
// 16-row A tile is staged in LDS (async copy) and shared by COLS/16 waves.
// ---------------------------------------------------------------------------
template<int K, int COLS, bool RELU, bool HAS_BIAS>
__global__ __launch_bounds__(32 * (COLS / 16))
void gemm_wmma_kernel(const float* __restrict__ A, const float* __restrict__ B,
                      const float* __restrict__ bias, float* __restrict__ C, int M)
{
    constexpr int NWAVES = COLS / 16;
    constexpr int KV     = K / 4;                 // float4s per A row
    __shared__ float As[16][K + 4];               // +4 float pad (16B-aligned rows,
                                                  //  bank-conflict-free column reads)
    const int rowTile = blockIdx.x * 16;

    // Stage the 16xK A tile into LDS with async b128 copies (clamp tail rows).
    for (int idx = threadIdx.x; idx < 16 * KV; idx += 32 * NWAVES) {
        int r  = idx / KV;
        int c4 = idx - r * KV;
        int row = rowTile + r;
        if (row >= M) row = M - 1;
        async_copy_b128_to_lds(&As[r][4 * c4], A + (size_t)row * K + 4 * c4);
    }
    async_copy_wait();
    __syncthreads();

    const int wave = threadIdx.x >> 5;
    const int lane = threadIdx.x & 31;
    const int grp  = lane >> 4;       // 0 -> K pair {0,1}; 1 -> K pair {2,3}
    const int lidx = lane & 15;
    const int col  = wave * 16 + lidx;

    v8f acc = {};
#pragma unroll 4
    for (int k0 = 0; k0 < K; k0 += 4) {
        const int ka = k0 + 2 * grp;
        v2f a, b;
        // A 16x4 layout: lanes 0-15 -> K={k0,k0+1}, lanes 16-31 -> K={k0+2,k0+3}
        a.x = As[lidx][ka];
        a.y = As[lidx][ka + 1];
        // B 4x16 layout: N striped across lanes within a VGPR, same K split
        b.x = B[(size_t)ka * COLS + col];
        b.y = B[(size_t)(ka + 1) * COLS + col];
        acc = __builtin_amdgcn_wmma_f32_16x16x4_f32(
            /*neg_a=*/false, a, /*neg_b=*/false, b,
            /*c_mod=*/(short)0, acc, /*reuse_a=*/false, /*reuse_b=*/false);
    }

    float bv = 0.0f;
    if (HAS_BIAS) bv = bias[col];

    // C/D layout: VGPR v -> row (rowTile + v + 8*grp), N = lidx.
    // Uniform full-tile test keeps the hot path free of per-lane EXEC churn.
    if (rowTile + 16 <= M) {
#pragma unroll
        for (int v = 0; v < 8; ++v) {
            const int row = rowTile + v + 8 * grp;
            float val = acc[v] + bv;
            if (RELU) val = fmaxf(val, 0.0f);
            C[(size_t)row * COLS + col] = val;
        }
    } else {
#pragma unroll
        for (int v = 0; v < 8; ++v) {
            const int row = rowTile + v + 8 * grp;
            if (row < M) {
                float val = acc[v] + bv;
                if (RELU) val = fmaxf(val, 0.0f);
                C[(size_t)row * COLS + col] = val;
            }
        }
    }
}

// ---------------------------------------------------------------------------
// Degree / normalization kernels (deg includes the self loop -> init to 1)
// ---------------------------------------------------------------------------
__global__ void init_deg_kernel(float* __restrict__ deg, int N)
{
    int i = blockIdx.x * blockDim.x + threadIdx.x;
    if (i < N) deg[i] = 1.0f;
}

__global__ void count_deg_kernel(const long long* __restrict__ dst,
                                 float* __restrict__ deg, int E)
{
    int e = blockIdx.x * blockDim.x + threadIdx.x;
    if (e < E) atomicAdd(&deg[(int)dst[e]], 1.0f);
}

__global__ void finish_dinv_kernel(float* __restrict__ deg, int N)
{
    int i = blockIdx.x * blockDim.x + threadIdx.x;
    if (i < N) deg[i] = __frsqrt_rn(deg[i]);   // deg >= 1 always (self loop)
}

// ---------------------------------------------------------------------------
// agg[i,:] = dinv[i]^2 * hW[i,:]   (self-loop message seeds the accumulator)
// One thread per float4 (32 float4s per node).
// ---------------------------------------------------------------------------
__global__ void self_init_kernel(const float* __restrict__ dinv,
                                 const float* __restrict__ hW,
                                 float* __restrict__ agg, int N)
{
    int t = blockIdx.x * blockDim.x + threadIdx.x;
    if (t >= N * (FDIM / 4)) return;
    const int node = t >> 5;
    const float s = dinv[node];
    const float n2 = s * s;
    float4 v = ((const float4*)hW)[t];
    v.x *= n2; v.y *= n2; v.z *= n2; v.w *= n2;
    ((float4*)agg)[t] = v;
}

// ---------------------------------------------------------------------------
// Edge scatter: one wave per edge (src/dst/norm wave-uniform).
// 32 lanes x float4 = 128 features; f32 global atomics into agg[dst].
// ---------------------------------------------------------------------------
__global__ void edge_scatter_kernel(const long long* __restrict__ src,
                                    const long long* __restrict__ dst,
                                    const float* __restrict__ dinv,
                                    const float* __restrict__ hW,
                                    float* __restrict__ agg, int E)
{
    int t = blockIdx.x * blockDim.x + threadIdx.x;
    int e = t >> 5;
    if (e >= E) return;
    const int lane = threadIdx.x & 31;
    const int s = (int)src[e];
    const int d = (int)dst[e];
    const float norm = dinv[s] * dinv[d];

    float4 m = ((const float4*)(hW + (size_t)s * FDIM))[lane];
    m.x *= norm; m.y *= norm; m.z *= norm; m.w *= norm;

    float* out = agg + (size_t)d * FDIM + lane * 4;
    atomicAdd(out + 0, m.x);
    atomicAdd(out + 1, m.y);
    atomicAdd(out + 2, m.z);
    atomicAdd(out + 3, m.w);
}

// ---------------------------------------------------------------------------
// h = ReLU(agg + bias), vectorized float4
// ---------------------------------------------------------------------------
__global__ void bias_relu_kernel(const float* __restrict__ agg,
                                 const float* __restrict__ bias,
                                 float* __restrict__ h, int N)
{
    int t = blockIdx.x * blockDim.x + threadIdx.x;
    if (t >= N * (FDIM / 4)) return;
    float4 a = ((const float4*)agg)[t];
    float4 b = ((const float4*)bias)[t & 31];
    a.x = fmaxf(a.x + b.x, 0.0f);
    a.y = fmaxf(a.y + b.y, 0.0f);
    a.z = fmaxf(a.z + b.z, 0.0f);
    a.w = fmaxf(a.w + b.w, 0.0f);
    ((float4*)h)[t] = a;
}

// ---------------------------------------------------------------------------
extern "C" void kernel_launch(void* const* d_in, const int* in_sizes, int n_in,
                              void* d_out, int out_size, void* d_ws, size_t ws_size,
                              hipStream_t stream)
{
    const float*     x     = (const float*)d_in[0];
    const long long* ei    = (const long long*)d_in[1];   // int64 [2, E]
    const float*     W_in  = (const float*)d_in[2];
    const float*     b_in  = (const float*)d_in[3];
    const float*     W_c1  = (const float*)d_in[4];
    const float*     b_c1  = (const float*)d_in[5];
    const float*     W_c2  = (const float*)d_in[6];
    const float*     b_c2  = (const float*)d_in[7];
    const float*     W_out = (const float*)d_in[8];
    const float*     b_out = (const float*)d_in[9];
    float*           out   = (float*)d_out;

    const int N = in_sizes[0] / 256;
    const int E = in_sizes[1] / 2;
    const long long* src = ei;
    const long long* dst = ei + E;

    // Workspace layout
    char* ws = (char*)d_ws;
    float* h    = (float*)ws;  ws += (size_t)N * FDIM * sizeof(float);
    float* tmp  = (float*)ws;  ws += (size_t)N * FDIM * sizeof(float);
    float* agg  = (float*)ws;  ws += (size_t)N * FDIM * sizeof(float);
    float* dinv = (float*)ws;

    const int rowTiles  = (N + 15) / 16;
    const int nodeVecs  = N * (FDIM / 4);            // float4 elements per feature map
    const int nodeGrid  = (nodeVecs + 255) / 256;
    const int edgeGrid  = ((E * 32) + 255) / 256;    // one wave per edge
    const int nGrid     = (N + 255) / 256;
    const int eGrid     = (E + 255) / 256;

    // Layer 1: h = ReLU(x @ W_in + b_in)
    gemm_wmma_kernel<256, 128, true, true><<<rowTiles, 256, 0, stream>>>(x, W_in, b_in, h, N);

    // Degree normalization (computed once, reused by both convs)
    init_deg_kernel<<<nGrid, 256, 0, stream>>>(dinv, N);
    count_deg_kernel<<<eGrid, 256, 0, stream>>>(dst, dinv, E);
    finish_dinv_kernel<<<nGrid, 256, 0, stream>>>(dinv, N);

    // GCNConv 1
    gemm_wmma_kernel<128, 128, false, false><<<rowTiles, 256, 0, stream>>>(h, W_c1, nullptr, tmp, N);
    self_init_kernel<<<nodeGrid, 256, 0, stream>>>(dinv, tmp, agg, N);
    edge_scatter_kernel<<<edgeGrid, 256, 0, stream>>>(src, dst, dinv, tmp, agg, E);
    bias_relu_kernel<<<nodeGrid, 256, 0, stream>>>(agg, b_c1, h, N);

    // GCNConv 2
    gemm_wmma_kernel<128, 128, false, false><<<rowTiles, 256, 0, stream>>>(h, W_c2, nullptr, tmp, N);
    self_init_kernel<<<nodeGrid, 256, 0, stream>>>(dinv, tmp, agg, N);
    edge_scatter_kernel<<<edgeGrid, 256, 0, stream>>>(src, dst, dinv, tmp, agg, E);
    bias_relu_kernel<<<nodeGrid, 256, 0, stream>>>(agg, b_c2, h, N);

    // Output layer: out = h @ W_out + b_out (no ReLU)
    gemm_wmma_kernel<128, 64, false, true><<<rowTiles, 128, 0, stream>>>(h, W_out, b_out, out, N);
}